// DigitCap_15479062135431
// MI455X (gfx1250) — compile-verified
//
#include <hip/hip_runtime.h>
#include <hip/hip_fp16.h>

// Problem constants (from reference): B=16, I=2048, A_IN=16, O=64, A=32
#define B_   16
#define I_   2048
#define K_   16
#define OA_  2048  // O*A

// d_ws layout (bytes) — requires ws_size >= 193,069,056 (~185 MiB):
//   votes  (fp16)  [B][I][OA]           :         0 .. 134217728
//   logits (f32)   [B][I][64]           : 134217728 .. 142606336
//   p1sum  (f32)   [128 chunks][B][OA]  : 142606336 .. 159383552
//   p35    (f32)   [B][256 waves][OA]   : 159383552 .. 192937984
//   act    (f32)   [B][OA]              : 192937984 .. 193069056

typedef float v2f __attribute__((ext_vector_type(2)));
typedef float v8f __attribute__((ext_vector_type(8)));

struct alignas(16) H8 { __half2 h[4]; };  // 8 fp16, one b128 load

// ---------------------------------------------------------------------------
// Pass 1: votes[b,i,:] = x[b,i,:16] @ W_i[16,2048] via V_WMMA_F32_16X16X4_F32,
// fused with column-sum partials for routing iteration 0 (uniform route).
// One wave per (n_tile 0..127, i_chunk 0..127); 4 WMMAs (K=16) per i.
// A layout (16x4 f32): lane<16 holds M=lane, v0=K0,v1=K1; lane>=16: K2,K3.
// C layout: vgpr r, lanes 0-15 -> M=r, lanes 16-31 -> M=r+8 (M = batch b).
// ---------------------------------------------------------------------------
__global__ void __launch_bounds__(256) k_votes(const float* __restrict__ x,
                                               const float* __restrict__ w,
                                               __half* __restrict__ votes,
                                               float* __restrict__ p1sum) {
  const int lane    = threadIdx.x & 31;
  const int wave    = threadIdx.x >> 5;
  const int gw      = blockIdx.x * 8 + wave;   // 16384 waves
  const int n_tile  = gw & 127;
  const int i_chunk = gw >> 7;
  const int n0      = n_tile * 16;
  const int m       = lane & 15;
  const int hi      = lane >> 4;

  v8f sum = {0.f,0.f,0.f,0.f,0.f,0.f,0.f,0.f};
  for (int ii = 0; ii < 16; ++ii) {
    const int i = i_chunk * 16 + ii;
    const float* xr = x + (size_t)m * (I_ * K_) + (size_t)i * K_;          // row M=m
    const float* wr = w + (size_t)i * (K_ * OA_) + n0 + m;                 // col N=m
    v8f c = {0.f,0.f,0.f,0.f,0.f,0.f,0.f,0.f};
#pragma unroll
    for (int s = 0; s < 4; ++s) {
      const int k = 4 * s + 2 * hi;
      v2f a;  a[0]  = xr[k];                 a[1]  = xr[k + 1];
      v2f bm; bm[0] = wr[(size_t)k * OA_];   bm[1] = wr[(size_t)(k + 1) * OA_];
      c = __builtin_amdgcn_wmma_f32_16x16x4_f32(false, a, false, bm,
                                                (short)0, c, false, false);
    }
#pragma unroll
    for (int r = 0; r < 8; ++r) {
      const int brow = r + 8 * hi;
      votes[(size_t)brow * ((size_t)I_ * OA_) + (size_t)i * OA_ + n0 + m] =
          __float2half(c[r]);
    }
    sum += c;
  }
#pragma unroll
  for (int r = 0; r < 8; ++r) {
    const int brow = r + 8 * hi;
    p1sum[((size_t)i_chunk * B_ + brow) * OA_ + n0 + m] = sum[r];
  }
}

// ---------------------------------------------------------------------------
// Fused routing pass (ITER=1: logits = d0; ITER=2: logits += d1):
// per (b,i): d[o] = sum_a votes*act;  route = softmax_o; accumulate
// route[o]*votes into a private per-wave slot (deterministic, no atomics).
// Grid: blockIdx = b*32 + chunk; 8 waves * 8 i each = 64 i per block.
// Lane owns o = 8j + lane/4 with a-range 8*(lane&3)..+7 (quad partition).
// ---------------------------------------------------------------------------
template <int ITER>
__global__ void __launch_bounds__(256) k_route(const __half* __restrict__ votes,
                                               const float* __restrict__ act,
                                               float* __restrict__ logits,
                                               float* __restrict__ pslots) {
  __shared__ float act_s[OA_];
  const int b     = blockIdx.x >> 5;
  const int chunk = blockIdx.x & 31;
  for (int idx = threadIdx.x; idx < OA_; idx += 256)
    act_s[idx] = act[b * OA_ + idx];
  __syncthreads();

  const int wave = threadIdx.x >> 5;
  const int lane = threadIdx.x & 31;
  float acc[8][8];
#pragma unroll
  for (int j = 0; j < 8; ++j)
#pragma unroll
    for (int q = 0; q < 8; ++q) acc[j][q] = 0.f;

  for (int it = 0; it < 8; ++it) {
    const int i = chunk * 64 + wave * 8 + it;
    const H8* vrow = (const H8*)(votes + ((size_t)b * I_ + i) * OA_);
    H8 pk[8];
    float p[8];
#pragma unroll
    for (int j = 0; j < 8; ++j) {
      pk[j] = vrow[j * 32 + lane];           // elements e0..e0+7, e0=256j+8*lane
      const int e0 = 256 * j + 8 * lane;
      float pj = 0.f;
#pragma unroll
      for (int q = 0; q < 4; ++q) {
        float2 f = __half22float2(pk[j].h[q]);
        pj += f.x * act_s[e0 + 2 * q] + f.y * act_s[e0 + 2 * q + 1];
      }
      p[j] = pj;
    }
#pragma unroll
    for (int j = 0; j < 8; ++j) {            // quad reduce -> full dot for o
      p[j] += __shfl_xor(p[j], 1, 32);
      p[j] += __shfl_xor(p[j], 2, 32);
    }
    const int ob = lane >> 2;                // o = 8j + ob
    const size_t lrow = ((size_t)b * I_ + i) * 64;
    float t[8];
#pragma unroll
    for (int j = 0; j < 8; ++j) {
      if (ITER == 1) {
        t[j] = p[j];
        if ((lane & 3) == 0) logits[lrow + 8 * j + ob] = p[j];
      } else {
        t[j] = p[j] + logits[lrow + 8 * j + ob];
      }
    }
    // softmax over the 64 o's (each o replicated in 4 lanes -> divide sum by 4)
    float mx = t[0];
#pragma unroll
    for (int j = 1; j < 8; ++j) mx = fmaxf(mx, t[j]);
#pragma unroll
    for (int msk = 1; msk < 32; msk <<= 1) mx = fmaxf(mx, __shfl_xor(mx, msk, 32));
    float ex[8], s = 0.f;
#pragma unroll
    for (int j = 0; j < 8; ++j) { ex[j] = __expf(t[j] - mx); s += ex[j]; }
#pragma unroll
    for (int msk = 1; msk < 32; msk <<= 1) s += __shfl_xor(s, msk, 32);
    const float inv = 4.0f / s;
#pragma unroll
    for (int j = 0; j < 8; ++j) {
      const float r = ex[j] * inv;
#pragma unroll
      for (int q = 0; q < 4; ++q) {
        float2 f = __half22float2(pk[j].h[q]);
        acc[j][2 * q]     += r * f.x;
        acc[j][2 * q + 1] += r * f.y;
      }
    }
  }
  float* slot = pslots + ((size_t)b * 256 + chunk * 8 + wave) * OA_;
#pragma unroll
  for (int j = 0; j < 8; ++j)
#pragma unroll
    for (int q = 0; q < 8; ++q)
      slot[256 * j + 8 * lane + q] = acc[j][q];
}

// ---------------------------------------------------------------------------
// Reduce partial slots, add bias, squash: out = t * ||t|| / (1 + ||t||^2).
// One wave per (b,o); lane = a.  mode 0: slots [s][B][OA]; mode 1: [B][ns][OA]
// ---------------------------------------------------------------------------
__global__ void __launch_bounds__(256) k_squash(const float* __restrict__ slots,
                                                const float* __restrict__ bias,
                                                float* __restrict__ out,
                                                float scale, int nslot, int mode) {
  const int gw   = blockIdx.x * 8 + (threadIdx.x >> 5);  // 1024 = 16b * 64o
  const int lane = threadIdx.x & 31;
  const int b = gw >> 6, o = gw & 63;
  const int e = o * 32 + lane;
  float t = 0.f;
  if (mode == 0) {
    for (int s = 0; s < nslot; ++s) t += slots[((size_t)s * B_ + b) * OA_ + e];
  } else {
    const float* sb = slots + (size_t)b * (size_t)nslot * OA_ + e;
    for (int s = 0; s < nslot; ++s) t += sb[(size_t)s * OA_];
  }
  t = t * scale + bias[e];
  float ns = t * t;
#pragma unroll
  for (int msk = 16; msk; msk >>= 1) ns += __shfl_xor(ns, msk, 32);
  const float norm = __fsqrt_rn(ns);
  out[b * OA_ + e] = t * norm / (1.f + ns);
}

// ---------------------------------------------------------------------------
extern "C" void kernel_launch(void* const* d_in, const int* in_sizes, int n_in,
                              void* d_out, int out_size, void* d_ws, size_t ws_size,
                              hipStream_t stream) {
  (void)in_sizes; (void)n_in; (void)out_size; (void)ws_size;
  const float* x    = (const float*)d_in[0];   // [16,2048,16,1]
  const float* w    = (const float*)d_in[1];   // [2048,16,2048]
  const float* bias = (const float*)d_in[2];   // [64,32]
  float* out = (float*)d_out;                  // [16,64,32]

  char* ws = (char*)d_ws;
  __half* votes = (__half*)(ws);
  float* logits = (float*)(ws + 134217728);
  float* p1     = (float*)(ws + 142606336);
  float* p35    = (float*)(ws + 159383552);
  float* act    = (float*)(ws + 192937984);

  // votes GEMM (WMMA f32 16x16x4) + iter-0 column sums
  k_votes<<<2048, 256, 0, stream>>>(x, w, votes, p1);
  // act0 = squash(sum/64 + bias)
  k_squash<<<128, 256, 0, stream>>>(p1, bias, act, 1.0f / 64.0f, 128, 0);
  // iter 0->1: d0, logits=d0, route, weighted sum
  k_route<1><<<512, 256, 0, stream>>>(votes, act, logits, p35);
  k_squash<<<128, 256, 0, stream>>>(p35, bias, act, 1.0f, 256, 1);
  // iter 1->2: d1, logits+d1, route, weighted sum (final)
  k_route<2><<<512, 256, 0, stream>>>(votes, act, logits, p35);
  k_squash<<<128, 256, 0, stream>>>(p35, bias, out, 1.0f, 256, 1);
}